// LIFSpike_66907000537576
// MI455X (gfx1250) — compile-verified
//
#include <hip/hip_runtime.h>

// ---------------------------------------------------------------------------
// LIF spike scan, [B=256, C=1024, T=128] f32 -> spikes f32 (same shape).
// Memory-bound (≈268 MB traffic, ~11.5 us floor at 23.3 TB/s on MI455X).
//
// Per 64-thread block (2 waves, wave32): one contiguous 32 KiB input tile.
//   load : TENSOR_LOAD_TO_LDS (TDM DMA, hardware row-padding into a
//          bank-skewed LDS layout, TENSORcnt)  [fallback: per-lane
//          GLOBAL_LOAD_ASYNC_TO_LDS_B128, ASYNCcnt]
//   scan : per-thread sequential LIF over its 132-float-stride LDS row,
//          spikes written in place (read-once stream)
//   drain: per-lane GLOBAL_STORE_ASYNC_FROM_LDS_B128, fully coalesced
// ---------------------------------------------------------------------------

#define AS1 __attribute__((address_space(1)))
#define AS3 __attribute__((address_space(3)))

typedef int          v4i   __attribute__((ext_vector_type(4)));
typedef unsigned int u32x4 __attribute__((ext_vector_type(4)));
typedef int          i32x8 __attribute__((ext_vector_type(8)));
typedef int          i32x4 __attribute__((ext_vector_type(4)));

#if __has_builtin(__builtin_amdgcn_global_load_async_to_lds_b128) && \
    __has_builtin(__builtin_amdgcn_global_store_async_from_lds_b128) && \
    __has_builtin(__builtin_amdgcn_s_wait_asynccnt)
#define HAVE_ASYNC_LDS 1
#else
#define HAVE_ASYNC_LDS 0
#endif

// tensor_load_to_lds arity differs by toolchain: 5 args on ROCm 7.2, 6 args
// on the toolchain that ships the TDM descriptor header.
#if HAVE_ASYNC_LDS && \
    __has_builtin(__builtin_amdgcn_tensor_load_to_lds) && \
    __has_builtin(__builtin_amdgcn_s_wait_tensorcnt)
#define HAVE_TDM 1
#if __has_include(<hip/amd_detail/amd_gfx1250_TDM.h>)
#define TDM_ARGS6 1
#else
#define TDM_ARGS6 0
#endif
#else
#define HAVE_TDM 0
#define TDM_ARGS6 0
#endif

namespace {
constexpr int   kT            = 128;               // time steps (innermost dim)
constexpr int   kThreads      = 64;                // 2 waves (wave32) per block
constexpr int   kRowsPerBlock = 64;                // one row per thread
constexpr int   kPadF         = 4;                 // 16B pad per row (bank skew)
constexpr int   kStrideF      = kT + kPadF;        // 132 floats
constexpr int   kStrideB      = kStrideF * 4;      // 528 bytes
constexpr int   kTileElems    = kRowsPerBlock * kT;          // 8192 floats
constexpr int   kCopyIters    = (kTileElems * 4) / 16 / kThreads;  // 32
constexpr float kTau          = 0.5f;
constexpr float kThresh       = 0.5f;
}  // namespace

// One LIF step: mem = mem*tau + x; spike = (mem - thresh > 0); mem -= spike*thresh.
__device__ __forceinline__ float lif_step(float& mem, float xi) {
  mem = fmaf(mem, kTau, xi);
  const float sp = (mem > kThresh) ? 1.0f : 0.0f;
  mem = fmaf(sp, -kThresh, mem);  // soft reset
  return sp;
}

__global__ __launch_bounds__(kThreads) void lif_scan_kernel(
    const float* __restrict__ x, float* __restrict__ out) {
  __shared__ __align__(16) float tile[kRowsPerBlock * kStrideF];

  const int tid = threadIdx.x;
  const size_t base = (size_t)blockIdx.x * (size_t)kTileElems;
  const float* gin = x + base;
  float* gout = out + base;

  AS3 char* lbase = (AS3 char*)(AS3 float*)tile;
  // Per-thread base offsets for the coalesced 16B-chunk mapping:
  //   chunk = it*64 + tid; global byte = tid*16 + it*1024;
  //   lds byte = (tid>>5)*528 + (tid&31)*16 + it*1056  (row-padded layout)
  const int lds_lane_off = (tid >> 5) * kStrideB + (tid & 31) * 16;

  // ---- Stage input tile: global -> LDS (row-padded) ----
#if HAVE_TDM
  if (tid < 32) {  // one TDM descriptor per block, issued by wave 0
    const unsigned lds_addr = (unsigned)(uintptr_t)lbase;
    const unsigned long long ga = (unsigned long long)(uintptr_t)gin;
    // D# group0: count=1 | lds_addr | global_addr[56:0] | type=2
    const u32x4 g0 = {1u,
                      lds_addr,
                      (unsigned)(ga & 0xFFFFFFFFu),
                      (unsigned)((ga >> 32) & 0x01FFFFFFu) | 0x80000000u};
    // D# group1: data_size=2 (4B) | pad_enable | pad_interval=6 (128 DWORDs
    // = one 512B row) | pad_amount=3 (4 DWORDs = 16B)  -> dw0 = 0x07920000.
    // 1-D tile: tensor_dim0 = tile_dim0 = 8192 elems, tensor_dim1 = 1,
    // tile_dim1 = 1, tensor_dim0_stride = 8192.
    const i32x8 g1 = {(int)0x07920000,
                      (int)((unsigned)kTileElems << 16),  // tensor_dim0[15:0]
                      (int)(1u << 16),                    // tensor_dim1 = 1
                      (int)((unsigned)kTileElems << 16),  // tile_dim0
                      1,                                  // tile_dim1 = 1
                      kTileElems,                         // tensor_dim0_stride
                      0, 0};
    const i32x4 g2 = {0, 0, 0, 0};
    const i32x4 g3 = {0, 0, 0, 0};
#if TDM_ARGS6
    const i32x8 g4 = {0, 0, 0, 0, 0, 0, 0, 0};
    __builtin_amdgcn_tensor_load_to_lds(g0, g1, g2, g3, g4, 0);
#else
    __builtin_amdgcn_tensor_load_to_lds(g0, g1, g2, g3, 0);
#endif
    __builtin_amdgcn_s_wait_tensorcnt(0);
  }
#elif HAVE_ASYNC_LDS
  {
    AS1 char* g = (AS1 char*)gin + tid * 16;
    AS3 char* l = lbase + lds_lane_off;
#pragma unroll 4
    for (int it = 0; it < kCopyIters; ++it) {
      __builtin_amdgcn_global_load_async_to_lds_b128((AS1 v4i*)g, (AS3 v4i*)l,
                                                     0, 0);
      g += kThreads * 16;   // 1024 B
      l += 2 * kStrideB;    // 1056 B (2 padded rows)
    }
    __builtin_amdgcn_s_wait_asynccnt(0);
  }
#else
#pragma unroll 4
  for (int it = 0; it < kCopyIters; ++it) {
    const int chunk = it * kThreads + tid;
    const int row = chunk >> 5;
    const int sub = chunk & 31;
    const float4 v = *reinterpret_cast<const float4*>(gin + (size_t)chunk * 4);
    *reinterpret_cast<float4*>(&tile[row * kStrideF + sub * 4]) = v;
  }
#endif
  __syncthreads();

  // ---- Per-row sequential LIF scan out of LDS; spikes written in place ----
  // Row stride 132 floats => lane L hits 16B granule (33*L + t) mod 16:
  // conflict-free per 16-lane phase for ds_load_b128/ds_store_b128.
  // unroll 8 bounds live ds_load results (~8 float4) to keep VGPR pressure
  // low enough that occupancy is LDS-bound, not VGPR-bound.
  {
    float mem = 0.0f;
    float4* rowp = reinterpret_cast<float4*>(&tile[tid * kStrideF]);
#pragma unroll 8
    for (int tc = 0; tc < kT / 4; ++tc) {
      const float4 v = rowp[tc];
      float4 s;
      s.x = lif_step(mem, v.x);
      s.y = lif_step(mem, v.y);
      s.z = lif_step(mem, v.z);
      s.w = lif_step(mem, v.w);
      rowp[tc] = s;  // overwrite input slot with spikes (read-once stream)
    }
  }
  __syncthreads();

  // ---- Drain spike tile: LDS -> global, fully coalesced ----
#if HAVE_ASYNC_LDS
  {
    AS1 char* g = (AS1 char*)gout + tid * 16;
    AS3 char* l = lbase + lds_lane_off;
#pragma unroll 4
    for (int it = 0; it < kCopyIters; ++it) {
      __builtin_amdgcn_global_store_async_from_lds_b128((AS1 v4i*)g,
                                                        (AS3 v4i*)l, 0, 0);
      g += kThreads * 16;   // 1024 B
      l += 2 * kStrideB;    // 1056 B
    }
    __builtin_amdgcn_s_wait_asynccnt(0);
  }
#else
#pragma unroll 4
  for (int it = 0; it < kCopyIters; ++it) {
    const int chunk = it * kThreads + tid;
    const int row = chunk >> 5;
    const int sub = chunk & 31;
    const float4 v = *reinterpret_cast<float4*>(&tile[row * kStrideF + sub * 4]);
    *reinterpret_cast<float4*>(gout + (size_t)chunk * 4) = v;
  }
#endif
}

extern "C" void kernel_launch(void* const* d_in, const int* in_sizes, int n_in,
                              void* d_out, int out_size, void* d_ws, size_t ws_size,
                              hipStream_t stream) {
  (void)n_in; (void)out_size; (void)d_ws; (void)ws_size;
  const float* x = (const float*)d_in[0];
  float* out = (float*)d_out;
  const int rows = in_sizes[0] / kT;              // B*C = 262144
  const int blocks = rows / kRowsPerBlock;        // 4096 blocks of 64 threads
  lif_scan_kernel<<<blocks, kThreads, 0, stream>>>(x, out);
}